// RoiPoolingConv_31851477467802
// MI455X (gfx1250) — compile-verified
//
#include <hip/hip_runtime.h>
#include <hip/hip_bf16.h>
#include <stdint.h>

// ROI bilinear pooling (ROI-align resize), MI455X / gfx1250.
// Store-bandwidth-bound: 411 MB out @ 23.3 TB/s ~ 18us floor; fm (164 MB)
// is L2-resident (192 MB L2). Strategy: async global->LDS staging of the 4
// corner channel-vectors per output pixel (double buffered, per-wave
// ASYNCcnt sync), non-temporal b128 output stores to preserve L2 residency.

#define POOL 14
#define FM_H 200
#define FM_W 200
#define FM_C 1024

typedef float v4f __attribute__((ext_vector_type(4)));

__device__ __forceinline__ uint32_t lds_u32(const void* p) {
  // generic pointer to __shared__ carries the LDS byte offset in its low 32 bits
  return (uint32_t)(uintptr_t)p;
}

__device__ __forceinline__ v4f lerp4(v4f a, v4f b, float t) {
  v4f r;
  r.x = __builtin_fmaf(b.x - a.x, t, a.x);
  r.y = __builtin_fmaf(b.y - a.y, t, a.y);
  r.z = __builtin_fmaf(b.z - a.z, t, a.z);
  r.w = __builtin_fmaf(b.w - a.w, t, a.w);
  return r;
}

// Exact IEEE fdiv that stays uniform:
//  1) "+v" asm barrier: no pass may rewrite x/14 into x*(1/14) (rounds
//     differently from the reference h.astype(f32)/POOL and can flip floor()
//     at integer crossings -> wrong gather row/col).
//  2) readfirstlane: legally moves the (uniform) quotient VGPR->SGPR so all
//     downstream index/address math re-scalarizes (SALU + clean s[n:n+1]
//     saddr operands for the async loads). A direct "+s" tie is an illegal
//     VGPR->SGPR copy since fdiv f32 is VALU-only.
__device__ __forceinline__ float exact_div_uniform(float a, float b) {
  float q = a / b;
  asm volatile("" : "+v"(q));
  int qi = __builtin_bit_cast(int, q);
  qi = __builtin_amdgcn_readfirstlane(qi);
  return __builtin_bit_cast(float, qi);
}

__global__ __launch_bounds__(256, 1) void roi_align_kernel(
    const float* __restrict__ fm, const int* __restrict__ rois,
    float* __restrict__ out) {
  const int tid = threadIdx.x;     // 0..255, handles channels [tid*4, tid*4+4)
  const int roi = blockIdx.x;      // 0..n_roi-1
  const int py  = blockIdx.y;      // 0..13

  // Block-uniform box -> scalar loads / SALU
  const int bx = rois[roi * 4 + 0];
  const int by = rois[roi * 4 + 1];
  const int bw = rois[roi * 4 + 2];
  const int bh = rois[roi * 4 + 3];

  // y interpolation (fixed for this block); exact h/POOL like the reference
  const float ys  = exact_div_uniform((float)bh, (float)POOL);
  const float sy  = (float)py * ys;
  const int   y0  = (int)sy;                 // sy >= 0 -> trunc == floor
  const float wy  = sy - (float)y0;
  const int   y0c = min(max(y0, 0), bh - 1);
  const int   y1c = min(max(y0 + 1, 0), bh - 1);
  const long long row0 = (long long)(by + y0c) * FM_W;
  const long long row1 = (long long)(by + y1c) * FM_W;

  const float xs = exact_div_uniform((float)bw, (float)POOL);

  // double buffer: [buf][corner][channel], 2*4*1024*4B = 32 KB
  __shared__ __align__(16) float lds[2][4][FM_C];

  const uint32_t voff = (uint32_t)(tid * 16);  // per-lane byte offset (b128)

  // Stage the 4 corner channel-vectors for pixel PX into LDS buffer BUF.
  // 4 async-to-LDS b128 instructions per wave => ASYNCcnt += 4.
#define STAGE(PX, BUF)                                                         \
  do {                                                                         \
    const float sx_  = (float)(PX)*xs;                                         \
    const int   x0_  = (int)sx_;                                               \
    const int   x0c_ = min(max(x0_, 0), bw - 1);                               \
    const int   x1c_ = min(max(x0_ + 1, 0), bw - 1);                           \
    const float* p00 = fm + (row0 + (bx + x0c_)) * (long long)FM_C;            \
    const float* p01 = fm + (row0 + (bx + x1c_)) * (long long)FM_C;            \
    const float* p10 = fm + (row1 + (bx + x0c_)) * (long long)FM_C;            \
    const float* p11 = fm + (row1 + (bx + x1c_)) * (long long)FM_C;            \
    asm volatile("global_load_async_to_lds_b128 %0, %1, %2"                    \
                 ::"v"(lds_u32(&lds[BUF][0][tid * 4])), "v"(voff), "s"(p00)    \
                 : "memory");                                                  \
    asm volatile("global_load_async_to_lds_b128 %0, %1, %2"                    \
                 ::"v"(lds_u32(&lds[BUF][1][tid * 4])), "v"(voff), "s"(p01)    \
                 : "memory");                                                  \
    asm volatile("global_load_async_to_lds_b128 %0, %1, %2"                    \
                 ::"v"(lds_u32(&lds[BUF][2][tid * 4])), "v"(voff), "s"(p10)    \
                 : "memory");                                                  \
    asm volatile("global_load_async_to_lds_b128 %0, %1, %2"                    \
                 ::"v"(lds_u32(&lds[BUF][3][tid * 4])), "v"(voff), "s"(p11)    \
                 : "memory");                                                  \
  } while (0)

  STAGE(0, 0);

#pragma unroll
  for (int px = 0; px < POOL; ++px) {
    const int cur = px & 1;
    if (px + 1 < POOL) {
      STAGE(px + 1, cur ^ 1);
      // allow the 4 just-issued next-stage loads to remain outstanding;
      // async loads complete in order, so current buffer is ready.
      asm volatile("s_wait_asynccnt 4" ::: "memory");
    } else {
      asm volatile("s_wait_asynccnt 0" ::: "memory");
    }

    const float sx = (float)px * xs;
    const int   x0 = (int)sx;
    const float wx = sx - (float)x0;

    const v4f g00 = *(const v4f*)&lds[cur][0][tid * 4];
    const v4f g01 = *(const v4f*)&lds[cur][1][tid * 4];
    const v4f g10 = *(const v4f*)&lds[cur][2][tid * 4];
    const v4f g11 = *(const v4f*)&lds[cur][3][tid * 4];

    const v4f top = lerp4(g00, g01, wx);
    const v4f bot = lerp4(g10, g11, wx);
    const v4f res = lerp4(top, bot, wy);

    // Non-temporal: keep the 411 MB output stream from evicting the
    // L2-resident feature map.
    const size_t o = ((size_t)(roi * POOL + py) * POOL + px) * FM_C +
                     (size_t)tid * 4;
    __builtin_nontemporal_store(res, (v4f*)(out + o));
  }
#undef STAGE
}

extern "C" void kernel_launch(void* const* d_in, const int* in_sizes, int n_in,
                              void* d_out, int out_size, void* d_ws,
                              size_t ws_size, hipStream_t stream) {
  const float* fm   = (const float*)d_in[0];  // (1,200,200,1024) f32
  const int*   rois = (const int*)d_in[1];    // (1,512,4) i32
  float*       out  = (float*)d_out;          // (1,512,14,14,1024) f32

  const int n_roi = in_sizes[1] / 4;
  dim3 grid(n_roi, POOL);
  roi_align_kernel<<<grid, 256, 0, stream>>>(fm, rois, out);
  (void)n_in; (void)out_size; (void)d_ws; (void)ws_size;
}